// CosineSimilarityLoss_83373905149952
// MI455X (gfx1250) — compile-verified
//
#include <hip/hip_runtime.h>
#include <math.h>

#define NUM_BINS 2000
#define PAD_BINS 2048        // pad to 32 * 64 for WMMA tiling; bins 2000..2047 stay zero
#define B_ROWS   2048
#define P_N      4096
#define T_N      4096

typedef __attribute__((ext_vector_type(2))) float v2f;
typedef __attribute__((ext_vector_type(8))) float v8f;

// One workgroup per batch row. 256 threads = 8 wave32s.
__global__ __launch_bounds__(256) void spec_cos_kernel(
    const float* __restrict__ pred_mz, const float* __restrict__ pred_in,
    const float* __restrict__ tgt_mz,  const float* __restrict__ tgt_in,
    const float* __restrict__ tgt_mask, float* __restrict__ row_cos)
{
    __shared__ float hp[PAD_BINS];
    __shared__ float ht[PAD_BINS];

    const int tid = threadIdx.x;
    const int row = blockIdx.x;

    // Zero both LDS histograms (8 entries per thread each).
    #pragma unroll
    for (int i = tid; i < PAD_BINS; i += 256) { hp[i] = 0.0f; ht[i] = 0.0f; }
    __syncthreads();

    const long base = (long)row * P_N;

    // ---- pred histogram: vectorized b128 loads, ds_add_f32 scatter ----
    {
        const float4* m4 = (const float4*)(pred_mz + base);
        const float4* x4 = (const float4*)(pred_in + base);
        #pragma unroll
        for (int v = tid; v < P_N / 4; v += 256) {
            float4 m = m4[v];
            float4 x = x4[v];
            int b0 = min(max((int)(m.x * 2000.0f), 0), NUM_BINS - 1);
            int b1 = min(max((int)(m.y * 2000.0f), 0), NUM_BINS - 1);
            int b2 = min(max((int)(m.z * 2000.0f), 0), NUM_BINS - 1);
            int b3 = min(max((int)(m.w * 2000.0f), 0), NUM_BINS - 1);
            atomicAdd(&hp[b0], x.x);
            atomicAdd(&hp[b1], x.y);
            atomicAdd(&hp[b2], x.z);
            atomicAdd(&hp[b3], x.w);
        }
    }

    // ---- target histogram (intensity * mask) ----
    {
        const float4* m4 = (const float4*)(tgt_mz + base);
        const float4* x4 = (const float4*)(tgt_in + base);
        const float4* k4 = (const float4*)(tgt_mask + base);
        #pragma unroll
        for (int v = tid; v < T_N / 4; v += 256) {
            float4 m = m4[v];
            float4 x = x4[v];
            float4 k = k4[v];
            int b0 = min(max((int)(m.x * 2000.0f), 0), NUM_BINS - 1);
            int b1 = min(max((int)(m.y * 2000.0f), 0), NUM_BINS - 1);
            int b2 = min(max((int)(m.z * 2000.0f), 0), NUM_BINS - 1);
            int b3 = min(max((int)(m.w * 2000.0f), 0), NUM_BINS - 1);
            atomicAdd(&ht[b0], x.x * k.x);
            atomicAdd(&ht[b1], x.y * k.y);
            atomicAdd(&ht[b2], x.z * k.z);
            atomicAdd(&ht[b3], x.w * k.w);
        }
    }
    __syncthreads();

    // ---- wave 0: WMMA row-sum reduction of S_pp, S_tt, S_pt over 2048 bins ----
    // V_WMMA_F32_16X16X4_F32 with B = ones: D[m][n] = sum_k A[m][k] + C[m][n].
    // 32 iterations * 64 elements (16 rows x K=4) cover all 2048 padded bins.
    if (tid < 32) {
        const int lane = tid;
        const int mrow = lane & 15;           // M for this lane
        const int kb   = (lane >> 4) << 1;    // K base: lanes 0-15 -> K=0,1; lanes 16-31 -> K=2,3

        v8f app = {};  // sum p^2
        v8f att = {};  // sum t^2
        v8f apt = {};  // sum p*t
        v2f ones; ones.x = 1.0f; ones.y = 1.0f;

        for (int it = 0; it < PAD_BINS / 64; ++it) {
            const int i0 = it * 64 + mrow * 4 + kb;
            float p0 = hp[i0], p1 = hp[i0 + 1];
            float t0 = ht[i0], t1 = ht[i0 + 1];

            v2f a;
            a.x = p0 * p0; a.y = p1 * p1;
            app = __builtin_amdgcn_wmma_f32_16x16x4_f32(
                false, a, false, ones, (short)0, app, false, false);
            a.x = t0 * t0; a.y = t1 * t1;
            att = __builtin_amdgcn_wmma_f32_16x16x4_f32(
                false, a, false, ones, (short)0, att, false, false);
            a.x = p0 * t0; a.y = p1 * t1;
            apt = __builtin_amdgcn_wmma_f32_16x16x4_f32(
                false, a, false, ones, (short)0, apt, false, false);
        }

        // Per-lane vertical sum of the 8 accumulator VGPRs.
        // C/D layout: VGPR r, lane n(0-15) = D[M=r][N=n]; lane 16+n = D[M=r+8][N=n].
        // => lane 0 holds column N=0 rows 0..7, lane 16 holds rows 8..15.
        float spp = 0.0f, stt = 0.0f, spt = 0.0f;
        #pragma unroll
        for (int r = 0; r < 8; ++r) { spp += app[r]; stt += att[r]; spt += apt[r]; }

        spp = __shfl(spp, 0, 32) + __shfl(spp, 16, 32);
        stt = __shfl(stt, 0, 32) + __shfl(stt, 16, 32);
        spt = __shfl(spt, 0, 32) + __shfl(spt, 16, 32);

        if (lane == 0) {
            float np = sqrtf(spp);
            float nt = sqrtf(stt);
            // dot of normalized spectra (norm + 1e-8 as in reference)
            float dotn = spt / ((np + 1e-8f) * (nt + 1e-8f));
            // norms of the normalized spectra, clamped by cosine eps
            float pn = fmaxf(np / (np + 1e-8f), 1e-8f);
            float tn = fmaxf(nt / (nt + 1e-8f), 1e-8f);
            row_cos[row] = dotn / (pn * tn);
        }
    }
}

// Deterministic fixed-order reduction of the 2048 per-row cosines.
__global__ __launch_bounds__(256) void finalize_kernel(
    const float* __restrict__ row_cos, float* __restrict__ out)
{
    __shared__ float s[256];
    const int tid = threadIdx.x;
    float acc = 0.0f;
    #pragma unroll
    for (int i = tid; i < B_ROWS; i += 256) acc += row_cos[i];
    s[tid] = acc;
    __syncthreads();
    #pragma unroll
    for (int off = 128; off > 0; off >>= 1) {
        if (tid < off) s[tid] += s[tid + off];
        __syncthreads();
    }
    if (tid == 0) out[0] = 1.0f - s[0] / (float)B_ROWS;
}

extern "C" void kernel_launch(void* const* d_in, const int* in_sizes, int n_in,
                              void* d_out, int out_size, void* d_ws, size_t ws_size,
                              hipStream_t stream) {
    const float* pred_mz  = (const float*)d_in[0];
    const float* pred_in  = (const float*)d_in[1];
    const float* tgt_mz   = (const float*)d_in[2];
    const float* tgt_in   = (const float*)d_in[3];
    const float* tgt_mask = (const float*)d_in[4];
    float* row_cos = (float*)d_ws;   // 2048 floats of scratch, fully rewritten each call
    float* out     = (float*)d_out;

    spec_cos_kernel<<<B_ROWS, 256, 0, stream>>>(
        pred_mz, pred_in, tgt_mz, tgt_in, tgt_mask, row_cos);
    finalize_kernel<<<1, 256, 0, stream>>>(row_cos, out);
}